// Norm_20710332301490
// MI455X (gfx1250) — compile-verified
//
#include <hip/hip_runtime.h>
#include <math.h>

typedef float v2f __attribute__((ext_vector_type(2)));
typedef float v4f __attribute__((ext_vector_type(4)));
typedef float v8f __attribute__((ext_vector_type(8)));

#define B_DIM 8
#define C_DIM 8
#define F_DIM 257
#define T_DIM 2000
#define EPSF  1e-6f

// ---------------------------------------------------------------------------
// Kernel 1: per-(batch, t) mean of |X[b,0,f,t]| over f=0..256 via WMMA.
// A = all-ones 16x4 fp32  =>  every row of D = column sums of the 4x16 B tile.
// One wave handles one (b, 16-wide t tile); 64 WMMAs cover f=0..255, f=256
// added with a scalar. Full fp32 precision (V_WMMA_F32_16X16X4_F32).
// ---------------------------------------------------------------------------
__global__ void colmean_wmma_kernel(const float* __restrict__ X,
                                    float* __restrict__ means) {
  const int lane = threadIdx.x & 31;
  const int wave = threadIdx.x >> 5;
  const int tile = blockIdx.x * (blockDim.x >> 5) + wave;   // 0..999 exactly
  const int b    = tile / 125;
  const int t0   = (tile % 125) * 16;
  const int col  = t0 + (lane & 15);
  const int khalf = (lane >> 4) << 1;                       // 0 or 2

  const float* base = X + (size_t)b * C_DIM * F_DIM * T_DIM;  // channel 0

  v2f a;
  a[0] = 1.0f;
  a[1] = 1.0f;
  v8f acc = {};

  for (int f0 = 0; f0 < 256; f0 += 4) {
    v2f bm;
    bm[0] = fabsf(base[(size_t)(f0 + khalf)     * T_DIM + col]);
    bm[1] = fabsf(base[(size_t)(f0 + khalf + 1) * T_DIM + col]);
    // D = A(ones) x B + C : accumulates column sums of each 4x16 tile.
    acc = __builtin_amdgcn_wmma_f32_16x16x4_f32(
        /*neg_a=*/false, a, /*neg_b=*/false, bm,
        /*c_mod=*/(short)0, acc, /*reuse_a=*/false, /*reuse_b=*/false);
  }

  // All rows of D are identical; acc[0] holds the sum for N = lane%16.
  float sum = acc[0] + fabsf(base[(size_t)256 * T_DIM + col]);
  if (lane < 16) {
    means[b * T_DIM + col] = sum * (1.0f / 257.0f);
  }
}

// ---------------------------------------------------------------------------
// Kernel 2: forgetting-normalization scan, parallelized as an affine prefix
// scan.  mu_t = a_t*mu_{t-1} + (1-a_t)*x_t  composes as (A,B) maps.
// One block per batch; 256 threads x 8 elements; Hillis-Steele in LDS.
// Overwrites means[] in place with the denominator (mu + eps).
// ---------------------------------------------------------------------------
__global__ void forgetting_scan_kernel(float* __restrict__ m) {
  const int b = blockIdx.x;
  const int j = threadIdx.x;
  const float alpha = 191.0f / 193.0f;
  float* row = m + b * T_DIM;

  const int t0 = j * 8;
  float x[8];
#pragma unroll
  for (int i = 0; i < 8; ++i) {
    const int t = t0 + i;
    x[i] = (t < T_DIM) ? row[t] : 0.0f;
  }

  // Compose this thread's 8-step affine map.
  float A = 1.0f, Bc = 0.0f;
#pragma unroll
  for (int i = 0; i < 8; ++i) {
    const int t = t0 + i;
    if (t < T_DIM) {
      const float tf = (float)t;
      const float at = fminf((tf - 1.0f) / (tf + 1.0f), alpha);
      A  = at * A;
      Bc = at * Bc + (1.0f - at) * x[i];
    }
  }

  __shared__ float sA[256];
  __shared__ float sB[256];
  sA[j] = A;
  sB[j] = Bc;
  __syncthreads();

  // Inclusive scan of affine maps: self ∘ prev.
  for (int off = 1; off < 256; off <<= 1) {
    float pA = 1.0f, pB = 0.0f;
    if (j >= off) { pA = sA[j - off]; pB = sB[j - off]; }
    __syncthreads();
    if (j >= off) {
      const float nA = sA[j] * pA;
      const float nB = sA[j] * pB + sB[j];
      sA[j] = nA;
      sB[j] = nB;
    }
    __syncthreads();
  }

  // Exclusive prefix (mu entering this segment); mu0 = 0 so only B matters.
  const float muStart = (j == 0) ? 0.0f : sB[j - 1];

  // Replay and write denominators in place.
  float mu = muStart;
#pragma unroll
  for (int i = 0; i < 8; ++i) {
    const int t = t0 + i;
    if (t < T_DIM) {
      const float tf = (float)t;
      const float at = fminf((tf - 1.0f) / (tf + 1.0f), alpha);
      mu = at * mu + (1.0f - at) * x[i];
      row[t] = mu + EPSF;
    }
  }
}

// ---------------------------------------------------------------------------
// Kernel 3: out = X / denom[b][t], B128 streaming with NT hints.
// T=2000 is divisible by 4, so every float4 stays within one t-row.
// denom is 64 KB and reused 2056x -> regular (cached) loads for it.
// ---------------------------------------------------------------------------
__global__ void normalize_kernel(const v4f* __restrict__ X4,
                                 const float* __restrict__ denom,
                                 v4f* __restrict__ out4,
                                 int total4) {
  const int idx = blockIdx.x * blockDim.x + threadIdx.x;
  if (idx >= total4) return;

  const int row4_per_b = C_DIM * F_DIM * (T_DIM / 4);   // 1,028,000
  const int b  = idx / row4_per_b;
  const int t4 = idx % (T_DIM / 4);

  const v4f r = *((const v4f*)denom + b * (T_DIM / 4) + t4);
  const v4f x = __builtin_nontemporal_load(X4 + idx);
  const v4f o = x / r;   // IEEE fp32 division, elementwise
  __builtin_nontemporal_store(o, out4 + idx);
}

// ---------------------------------------------------------------------------
extern "C" void kernel_launch(void* const* d_in, const int* in_sizes, int n_in,
                              void* d_out, int out_size, void* d_ws, size_t ws_size,
                              hipStream_t stream) {
  const float* X  = (const float*)d_in[0];
  float* out      = (float*)d_out;
  float* means    = (float*)d_ws;            // 8*2000 floats = 64 KB scratch

  // 1) column means via WMMA: 1000 tiles, 4 waves/block -> 250 blocks.
  colmean_wmma_kernel<<<250, 128, 0, stream>>>(X, means);

  // 2) forgetting-normalization scan, one block per batch (in-place -> denom).
  forgetting_scan_kernel<<<B_DIM, 256, 0, stream>>>(means);

  // 3) streaming elementwise divide.
  const int total4 = B_DIM * C_DIM * F_DIM * (T_DIM / 4); // 8,224,000
  const int blocks = (total4 + 255) / 256;
  normalize_kernel<<<blocks, 256, 0, stream>>>((const v4f*)X, means,
                                               (v4f*)out, total4);
}